// SimpleEnergyModel_69810398429869
// MI455X (gfx1250) — compile-verified
//
#include <hip/hip_runtime.h>

typedef float v2f __attribute__((ext_vector_type(2)));
typedef float v8f __attribute__((ext_vector_type(8)));

#define NBLK 64
#define NTHR 256

// Kernel 1: per-thread accumulation of (x, y, z, |x|^2), then a wave-level
// reduction of the 256 per-thread partials done on the WMMA pipe:
//   C = A(16x4, data) * B(4x16, ones)  =>  C[m,n] = sum_k A[m,k]
// Row classes: rows 0-3 = X, 4-7 = Y, 8-11 = Z, 12-15 = |x|^2.
// A layout (ISA 7.12.2, 32-bit A 16x4): row m <- lane m {v0=K0, v1=K1} and
// lane m+16 {v0=K2, v1=K3}.  B = all ones, so B's layout is irrelevant.
__global__ void energy_partial_kernel(const float* __restrict__ pos, int npts,
                                      float* __restrict__ ws) {
    __shared__ float lds[4][NTHR];   // SoA: [component][thread]
    const int tid    = threadIdx.x;
    const int gtid   = blockIdx.x * blockDim.x + tid;
    const int stride = gridDim.x * blockDim.x;

    float sx = 0.f, sy = 0.f, sz = 0.f, s2 = 0.f;
    for (int i = gtid; i < npts; i += stride) {
        const float x = pos[3 * i + 0];
        const float y = pos[3 * i + 1];
        const float z = pos[3 * i + 2];
        sx += x; sy += y; sz += z;
        s2 += x * x + y * y + z * z;
    }
    lds[0][tid] = sx; lds[1][tid] = sy; lds[2][tid] = sz; lds[3][tid] = s2;
    __syncthreads();

    // Every wave does the identical WMMA reduction (EXEC all-ones per wave;
    // results are the same in all waves, thread 0 writes).
    const int L   = tid & 31;                 // lane in wave32
    const int row = L & 15;                   // A-matrix row this lane feeds
    const int cls = row >> 2;                 // 0=X 1=Y 2=Z 3=|x|^2
    const int wi  = (row & 3) + ((L >> 4) << 2);  // 0..7 within class

    v8f c = {0.f, 0.f, 0.f, 0.f, 0.f, 0.f, 0.f, 0.f};
    v2f b = {1.0f, 1.0f};                     // ones: layout-independent
#pragma unroll
    for (int t = 0; t < 16; ++t) {            // 16 WMMAs reduce 256/component
        v2f a;
        a.x = lds[cls][t * 16 + 2 * wi + 0];
        a.y = lds[cls][t * 16 + 2 * wi + 1];
        c = __builtin_amdgcn_wmma_f32_16x16x4_f32(
                /*neg_a=*/false, a, /*neg_b=*/false, b,
                /*c_mod=*/(short)0, c, /*reuse_a=*/false, /*reuse_b=*/false);
    }
    // C/D layout: VGPR r holds row r (lanes 0-15) and row r+8 (lanes 16-31).
    const float sumA = c[0] + c[1] + c[2] + c[3]; // lanes0-15: SX, lanes16-31: SZ
    const float sumB = c[4] + c[5] + c[6] + c[7]; // lanes0-15: SY, lanes16-31: S2
    const float SX = __shfl(sumA, 0, 32);
    const float SZ = __shfl(sumA, 16, 32);
    const float SY = __shfl(sumB, 0, 32);
    const float SS = __shfl(sumB, 16, 32);

    if (tid == 0) {
        ws[blockIdx.x * 4 + 0] = SX;
        ws[blockIdx.x * 4 + 1] = SY;
        ws[blockIdx.x * 4 + 2] = SZ;
        ws[blockIdx.x * 4 + 3] = SS;
    }
}

// Kernel 2: one wave reduces the 64 block partials deterministically and
// applies the closed form  E = 2*N*sum|x|^2 - 2*||sum x||^2.
__global__ void energy_final_kernel(const float* __restrict__ ws,
                                    float* __restrict__ out,
                                    int nblk, int npts) {
    const int lane = threadIdx.x;  // 32 threads = 1 wave
    float ax = 0.f, ay = 0.f, az = 0.f, a2 = 0.f;
    for (int i = lane; i < nblk; i += 32) {
        ax += ws[i * 4 + 0];
        ay += ws[i * 4 + 1];
        az += ws[i * 4 + 2];
        a2 += ws[i * 4 + 3];
    }
    for (int off = 16; off > 0; off >>= 1) {
        ax += __shfl_down(ax, off, 32);
        ay += __shfl_down(ay, off, 32);
        az += __shfl_down(az, off, 32);
        a2 += __shfl_down(a2, off, 32);
    }
    if (lane == 0) {
        const float energy =
            2.0f * (float)npts * a2 - 2.0f * (ax * ax + ay * ay + az * az);
        out[0] = energy;
    }
}

extern "C" void kernel_launch(void* const* d_in, const int* in_sizes, int n_in,
                              void* d_out, int out_size, void* d_ws, size_t ws_size,
                              hipStream_t stream) {
    (void)n_in; (void)out_size; (void)ws_size;
    const float* pos = (const float*)d_in[0];
    const int npts = in_sizes[0] / 3;          // [N,3] float32, flat count
    float* ws  = (float*)d_ws;                 // 64 blocks * 4 floats = 1 KB
    float* out = (float*)d_out;                // single float32

    energy_partial_kernel<<<NBLK, NTHR, 0, stream>>>(pos, npts, ws);
    energy_final_kernel<<<1, 32, 0, stream>>>(ws, out, NBLK, npts);
}